// PhoneticPriorModel_63419487093311
// MI455X (gfx1250) — compile-verified
//
#include <hip/hip_runtime.h>

#define K_PH   40
#define F_DIM  64
#define LC     40
#define VOCAB  512
#define WRD    8
#define NLEN   96
#define BATCH  2
#define NSPAN  7
#define ALPHA_C 3.5f
#define BIGF   1000000000.0f

typedef __attribute__((ext_vector_type(2))) float v2f;
typedef __attribute__((ext_vector_type(8))) float v8f;

// ---- WMMA helpers: C(16x16) += A(16x4) x B(4x16) over kdim, f32 -------------
// A layout (ISA 7.12.2): lanes 0-15 hold M=lane; VGPR0=K(2*half+0), VGPR1=K(2*half+1)
// B assumed mirrored with N in place of M.
__device__ inline v8f wmma_k4(const float* __restrict__ A, int lda,
                              const float* __restrict__ B, int ldb,
                              int mbase, int nbase, int kdim, bool transB,
                              v8f acc) {
  const int lane = threadIdx.x & 31;
  const int half = lane >> 4;
  const int l15  = lane & 15;
  for (int k0 = 0; k0 < kdim; k0 += 4) {
    const int ka = k0 + 2 * half;
    v2f a, b;
    a.x = A[(mbase + l15) * lda + ka];
    a.y = A[(mbase + l15) * lda + ka + 1];
    if (!transB) {
      b.x = B[ka * ldb + (nbase + l15)];
      b.y = B[(ka + 1) * ldb + (nbase + l15)];
    } else {
      b.x = B[(nbase + l15) * ldb + ka];
      b.y = B[(nbase + l15) * ldb + ka + 1];
    }
    acc = __builtin_amdgcn_wmma_f32_16x16x4_f32(false, a, false, b,
                                                (short)0, acc, false, false);
  }
  return acc;
}

__device__ inline void store_tile(float* __restrict__ C, int ldc,
                                  int mbase, int nbase, v8f acc) {
  const int lane = threadIdx.x & 31;
  const int half = lane >> 4;
  const int l15  = lane & 15;
#pragma unroll
  for (int r = 0; r < 8; ++r)
    C[(mbase + r + 8 * half) * ldc + nbase + l15] = acc[r];
}

// ---- Kernel A: GEMM chain + softmaxes + logP + omega_loss -------------------
__global__ void __launch_bounds__(256)
phon_stage_a(const float* __restrict__ ipa, const float* __restrict__ proj,
             const float* __restrict__ ml, float* __restrict__ ws_logP,
             float* __restrict__ ws_omega) {
  __shared__ float sM[48 * 64];  // ipa padded -> later mixT (stride 48)
  __shared__ float sN[64 * 64];  // proj^T -> later char_distr (stride 40)
  __shared__ float sK[48 * 64];  // k_emb
  __shared__ float sL[48 * 64];  // l_emb
  __shared__ float sS[48 * 48];  // mapping_logits -> scores
  __shared__ float sMass[LC];
  const int tid = threadIdx.x;
  const int wave = tid >> 5;

  for (int i = tid; i < 48 * 64; i += 256) sM[i] = 0.0f;
  __syncthreads();
  for (int i = tid; i < K_PH * F_DIM; i += 256)
    sM[(i / F_DIM) * 64 + (i % F_DIM)] = ipa[i];
  for (int i = tid; i < 64 * 64; i += 256) {  // sN[f][dd] = proj[dd][f]
    int dd = i >> 6, f = i & 63;
    sN[f * 64 + dd] = proj[i];
  }
  for (int i = tid; i < K_PH * LC; i += 256)
    sS[(i / LC) * 48 + (i % LC)] = ml[i];
  __syncthreads();

  // k_emb(40x64) = ipa(40x64) x projT(64x64)
  for (int t = wave; t < 12; t += 8) {
    int mt = t >> 2, nt = t & 3;
    v8f acc = {};
    acc = wmma_k4(sM, 64, sN, 64, mt * 16, nt * 16, 64, false, acc);
    store_tile(sK, 64, mt * 16, nt * 16, acc);
  }
  __syncthreads();

  // mixT (softmax over K per column l) into zeroed sM (stride 48)
  for (int i = tid; i < 48 * 64; i += 256) sM[i] = 0.0f;
  __syncthreads();
  if (tid < LC) {
    const int l = tid;
    float mx = -1e30f;
    for (int k = 0; k < K_PH; ++k) mx = fmaxf(mx, sS[k * 48 + l]);
    float sum = 0.f;
    for (int k = 0; k < K_PH; ++k) sum += __expf(sS[k * 48 + l] - mx);
    const float inv = 1.0f / sum;
    for (int k = 0; k < K_PH; ++k)
      sM[l * 48 + k] = __expf(sS[k * 48 + l] - mx) * inv;
  }
  __syncthreads();

  // l_emb(40x64) = mixT(40x40) x k_emb(40x64)
  for (int t = wave; t < 12; t += 8) {
    int mt = t >> 2, nt = t & 3;
    v8f acc = {};
    acc = wmma_k4(sM, 48, sK, 64, mt * 16, nt * 16, 40, false, acc);
    store_tile(sL, 64, mt * 16, nt * 16, acc);
  }
  __syncthreads();

  // scores(40x40) = k_emb x l_emb^T
  for (int t = wave; t < 9; t += 8) {
    int mt = t / 3, nt = t % 3;
    v8f acc = {};
    acc = wmma_k4(sK, 64, sL, 64, mt * 16, nt * 16, 64, true, acc);
    store_tile(sS, 48, mt * 16, nt * 16, acc);
  }
  __syncthreads();

  // char_distr = row softmax(scores*5); logP = log(cd + 1e-9)
  if (tid < K_PH) {
    const int k = tid;
    float mx = -1e30f;
    for (int l = 0; l < LC; ++l) mx = fmaxf(mx, sS[k * 48 + l] * 5.0f);
    float sum = 0.f;
    for (int l = 0; l < LC; ++l) sum += __expf(sS[k * 48 + l] * 5.0f - mx);
    const float inv = 1.0f / sum;
    for (int l = 0; l < LC; ++l) {
      float cd = __expf(sS[k * 48 + l] * 5.0f - mx) * inv;
      sN[k * LC + l] = cd;
      ws_logP[k * LC + l] = __logf(cd + 1e-9f);
    }
  }
  __syncthreads();
  if (tid < LC) {
    float m = 0.f;
    for (int k = 0; k < K_PH; ++k) m += sN[k * LC + tid];
    sMass[tid] = m;
  }
  __syncthreads();
  if (tid == 0) {
    float target = 0.f;
    for (int l = 0; l < LC; ++l) target += sMass[l];
    target *= (1.0f / LC);
    float ol = 0.f;
    for (int l = 0; l < LC; ++l) {
      float d = sMass[l] - target;
      ol += d * d;
    }
    ws_omega[0] = ol * (1.0f / LC);
  }
}

// ---- Kernel B: soft edit DP, one thread per vocab word, 7 span lengths -----
__global__ void __launch_bounds__(512)
phon_edit_dp(const int* __restrict__ insc, const int* __restrict__ vocab,
             const float* __restrict__ ws_logP, float* __restrict__ ws_pspan) {
  __shared__ float sLogP[K_PH * LC];
  __shared__ float red[VOCAB];
  __shared__ int sid10[10];
  const int tid = threadIdx.x;
  const int b = blockIdx.x / NLEN;
  const int n = blockIdx.x % NLEN;

  for (int i = tid; i < K_PH * LC; i += VOCAB) sLogP[i] = ws_logP[i];
  if (tid < 10) {
    int p = n + 1 - 10 + tid;
    p = max(0, min(NLEN - 1, p));
    sid10[tid] = insc[b * NLEN + p];
  }
  int kid[WRD];
#pragma unroll
  for (int c = 0; c < WRD; ++c) kid[c] = vocab[tid * WRD + c];
  __syncthreads();

  for (int li = 0; li < NSPAN; ++li) {
    const int l = li + 4;
    float row[WRD + 1];
#pragma unroll
    for (int j = 0; j <= WRD; ++j) row[j] = ALPHA_C * (float)j;
    for (int r = 0; r < l; ++r) {
      const int sid = sid10[10 - l + r];
      float diag = row[0];
      float d1 = row[0] + ALPHA_C;
      float d2 = BIGF;
      row[0] = d1;
#pragma unroll
      for (int j = 1; j <= WRD; ++j) {
        const float up = row[j];
        const float s = -sLogP[kid[j - 1] * LC + sid];
        const float c0 = up + ALPHA_C;
        const float c1 = d1 + ALPHA_C;
        const float c2 = diag + s;
        const float c3 = (j >= 2) ? (d2 + ALPHA_C) : BIGF;
        const float m = fminf(fminf(c0, c1), fminf(c2, c3));
        const float val =
            m - __logf(__expf(m - c0) + __expf(m - c1) + __expf(m - c2) +
                       __expf(m - c3));
        d2 = d1; d1 = val; diag = up; row[j] = val;
      }
    }
    const float sc = -row[WRD];

    // block logsumexp over V=512
    red[tid] = sc;
    __syncthreads();
    for (int s2 = VOCAB / 2; s2 > 0; s2 >>= 1) {
      if (tid < s2) red[tid] = fmaxf(red[tid], red[tid + s2]);
      __syncthreads();
    }
    const float mx = red[0];
    __syncthreads();
    red[tid] = __expf(sc - mx);
    __syncthreads();
    for (int s2 = VOCAB / 2; s2 > 0; s2 >>= 1) {
      if (tid < s2) red[tid] += red[tid + s2];
      __syncthreads();
    }
    if (tid == 0) {
      float ps = mx + __logf(red[0]) - 6.2383246250395075f;  // log(512)
      if (n + 1 < l) ps = -BIGF;
      ws_pspan[(b * NLEN + n) * NSPAN + li] = ps;
    }
    __syncthreads();
  }
}

// ---- Kernel C: sequential segmentation DP + final scalars ------------------
__global__ void __launch_bounds__(32)
phon_seg_dp(const float* __restrict__ ws_pspan,
            const float* __restrict__ ws_omega, float* __restrict__ out) {
  __shared__ float q[BATCH], cv[BATCH];
  const int tid = threadIdx.x;
  if (tid < BATCH) {
    float dp[10], cov[10];
    dp[0] = 0.f;
    for (int j = 1; j < 10; ++j) dp[j] = -BIGF;
    for (int j = 0; j < 10; ++j) cov[j] = 0.f;
    const float log_p_o = __logf(0.2f);
    const float log_p_el = __logf(0.8f / 7.0f);
    for (int n = 0; n < NLEN; ++n) {
      const float* ps = &ws_pspan[(tid * NLEN + n) * NSPAN];
      const float c_o = dp[0] + log_p_o;
      float cel[NSPAN];
      float mx = c_o;
      for (int k = 0; k < NSPAN; ++k) {
        cel[k] = dp[k + 3] + log_p_el + ps[k];
        mx = fmaxf(mx, cel[k]);
      }
      float sum = __expf(c_o - mx);
      for (int k = 0; k < NSPAN; ++k) sum += __expf(cel[k] - mx);
      const float dp_i = mx + __logf(sum);
      int chosen = 0;
      for (int k = 0; k < NSPAN; ++k)
        if (cel[k] >= c_o) chosen = k + 4;
      const int ci = (chosen > 0 ? chosen : 1) - 1;
      const float cov_pick = cov[ci] + (float)chosen;
      const float cov_i = fminf(chosen > 0 ? cov_pick : cov[0], (float)NLEN);
      for (int j = 9; j >= 1; --j) { dp[j] = dp[j - 1]; cov[j] = cov[j - 1]; }
      dp[0] = dp_i;
      cov[0] = cov_i;
    }
    q[tid] = dp[0];
    cv[tid] = cov[0] * (1.0f / NLEN);
  }
  __syncthreads();
  if (tid == 0) {
    const float quality = 0.5f * (q[0] + q[1]);
    const float ocov = 0.5f * (cv[0] + cv[1]);
    const float oloss = ws_omega[0];
    out[0] = quality + 10.0f * ocov - 100.0f * oloss;
    out[1] = quality;
    out[2] = ocov;
    out[3] = oloss;
  }
}

extern "C" void kernel_launch(void* const* d_in, const int* in_sizes, int n_in,
                              void* d_out, int out_size, void* d_ws,
                              size_t ws_size, hipStream_t stream) {
  (void)in_sizes; (void)n_in; (void)out_size; (void)ws_size;
  const int* insc = (const int*)d_in[0];       // (2,96) int32
  const int* vocab = (const int*)d_in[1];      // (512,8) int32
  const float* ipa = (const float*)d_in[2];    // (40,64) f32
  const float* proj = (const float*)d_in[3];   // (64,64) f32
  const float* ml = (const float*)d_in[4];     // (40,40) f32
  float* out = (float*)d_out;                  // 4 f32 scalars
  float* ws = (float*)d_ws;
  float* ws_logP = ws;                         // 1600 floats
  float* ws_omega = ws + 1600;                 // 1 float
  float* ws_pspan = ws + 1664;                 // 2*96*7 = 1344 floats

  phon_stage_a<<<1, 256, 0, stream>>>(ipa, proj, ml, ws_logP, ws_omega);
  phon_edit_dp<<<BATCH * NLEN, VOCAB, 0, stream>>>(insc, vocab, ws_logP,
                                                   ws_pspan);
  phon_seg_dp<<<1, 32, 0, stream>>>(ws_pspan, ws_omega, out);
}